// AnEn_56547539419656
// MI455X (gfx1250) — compile-verified
//
#include <hip/hip_runtime.h>
#include <hip/hip_bf16.h>

typedef __attribute__((ext_vector_type(2))) float v2f;
typedef __attribute__((ext_vector_type(4))) float v4f;
typedef __attribute__((ext_vector_type(8))) float v8f;

#define NF      8
#define C_ROWS  510
#define H_ROWS  9998
#define CPAD    512
#define HPAD    10000
#define CT_TILES 32          // ceil(510/16)
#define HT_TILES 625         // ceil(9998/16), covers 10000 cols
#define N_TILES (CT_TILES * HT_TILES)   // 20000
#define KSEL    50

// workspace layout in floats
#define WS_WNORM 0
#define WS_SA    8                       // [8][512]   feature-major
#define WS_SB    (8 + NF * CPAD)         // [10000][8] row-major
#define WS_D     84112                   // [510][10000], 16B aligned

__device__ __forceinline__ int imin(int a, int b) { return a < b ? a : b; }

// ---------------------------------------------------------------------------
// Kernel 1: per-feature std over x_h -> w_norm = fw / max(std, 1e-8)
// ---------------------------------------------------------------------------
__global__ void anen_stats(const float* __restrict__ xh,
                           const float* __restrict__ fw,
                           float* __restrict__ wnorm) {
    __shared__ float s_sum[256 * NF];
    __shared__ float s_sq [256 * NF];
    const int t = threadIdx.x;
    float sum[NF], sq[NF];
#pragma unroll
    for (int f = 0; f < NF; ++f) { sum[f] = 0.f; sq[f] = 0.f; }
    for (int r = t; r < 10000; r += 256) {
#pragma unroll
        for (int f = 0; f < NF; ++f) {
            float v = xh[r * NF + f];
            sum[f] += v;
            sq[f]  += v * v;
        }
    }
#pragma unroll
    for (int f = 0; f < NF; ++f) { s_sum[t * NF + f] = sum[f]; s_sq[t * NF + f] = sq[f]; }
    __syncthreads();
    for (int s = 128; s > 0; s >>= 1) {
        if (t < s) {
#pragma unroll
            for (int f = 0; f < NF; ++f) {
                s_sum[t * NF + f] += s_sum[(t + s) * NF + f];
                s_sq [t * NF + f] += s_sq [(t + s) * NF + f];
            }
        }
        __syncthreads();
    }
    if (t < NF) {
        float mean = s_sum[t] * (1.0f / 10000.f);
        float var  = s_sq[t] * (1.0f / 10000.f) - mean * mean;
        var = fmaxf(var, 0.f);
        float sd = fmaxf(sqrtf(var), 1e-8f);
        wnorm[t] = fw[t] / sd;
    }
}

// ---------------------------------------------------------------------------
// Kernel 2: windowed sum-of-squares
//   val(c,f) = sum_{w=0..2} X[min(c+w, n-1)][f]^2
//   fmajor=1 -> S[f*cpad + c]   (sa layout)
//   fmajor=0 -> S[c*NF + f]     (sb layout)
// ---------------------------------------------------------------------------
__global__ void anen_wsq(const float* __restrict__ X,
                         float* __restrict__ S, int n, int cpad, int fmajor) {
    int idx = blockIdx.x * blockDim.x + threadIdx.x;
    if (idx >= NF * cpad) return;
    int f = idx / cpad;
    int c = idx - f * cpad;
    float s = 0.f;
#pragma unroll
    for (int w = 0; w < 3; ++w) {
        int r = imin(c + w, n - 1);
        float v = X[r * NF + f];
        s = fmaf(v, v, s);
    }
    if (fmajor) S[f * cpad + c] = s;
    else        S[c * NF + f]   = s;
}

// ---------------------------------------------------------------------------
// Kernel 3: WMMA distance tiles. One wave = one 16x16 (c,h) tile.
// Per feature f: cross[c,h] = sum_{w<3} xt[c+w,f]*xh[h+w,f] via
// V_WMMA_F32_16X16X4_F32 (K padded to 4 with zeros, branch-free), then
//   dist += wnorm[f] * sqrt(max(sa + sb - 2*cross, 0))
// ---------------------------------------------------------------------------
__global__ void __launch_bounds__(128)
anen_dist_wmma(const float* __restrict__ xt,   // [512][8]
               const float* __restrict__ xh,   // [10000][8]
               const float* __restrict__ wnorm,// [8]
               const float* __restrict__ sa,   // [8][512]
               const float* __restrict__ sbr,  // [10000][8]
               float* __restrict__ D)          // [510][10000]
{
    const int wave = threadIdx.x >> 5;
    const int lane = threadIdx.x & 31;
    const int tile = blockIdx.x * 4 + wave;      // grid 5000 * 4 waves = 20000
    const int ct = tile / HT_TILES;
    const int ht = tile - ct * HT_TILES;

    const int half = lane >> 4;                  // 0: K=0/1, 1: K=2/3
    const int l16  = lane & 15;
    const int k0   = half * 2;

    const int cm    = ct * 16 + l16;             // A-matrix row M for this lane
    const int hn    = ht * 16 + l16;             // B-matrix col N / C col
    const int cbase = ct * 16 + half * 8;        // first C-row this lane owns

    const int cmk0 = imin(cm + k0, CPAD - 1);
    const int cm1  = imin(cm + 1,  CPAD - 1);
    const int hnk0 = imin(hn + k0, HPAD - 1);
    const int hn1  = imin(hn + 1,  HPAD - 1);

    const float pad = half ? 0.0f : 1.0f;        // zero K=3 without branches

    // Whole-row vector loads: each row is 8 floats, 32B aligned.
    const v4f* pA0 = (const v4f*)(xt + cmk0 * NF);
    const v4f* pA1 = (const v4f*)(xt + cm1  * NF);
    const v4f* pB0 = (const v4f*)(xh + hnk0 * NF);
    const v4f* pB1 = (const v4f*)(xh + hn1  * NF);
    const v4f* pSB = (const v4f*)(sbr + hn  * NF);
    const v4f* pWN = (const v4f*)(wnorm);
    v4f A0l = pA0[0], A0h = pA0[1];
    v4f A1l = pA1[0], A1h = pA1[1];
    v4f B0l = pB0[0], B0h = pB0[1];
    v4f B1l = pB1[0], B1h = pB1[1];
    v4f SBl = pSB[0], SBh = pSB[1];
    v4f WNl = pWN[0], WNh = pWN[1];

    float a0[NF] = { A0l.x, A0l.y, A0l.z, A0l.w, A0h.x, A0h.y, A0h.z, A0h.w };
    float a1[NF] = { A1l.x, A1l.y, A1l.z, A1l.w, A1h.x, A1h.y, A1h.z, A1h.w };
    float b0[NF] = { B0l.x, B0l.y, B0l.z, B0l.w, B0h.x, B0h.y, B0h.z, B0h.w };
    float b1[NF] = { B1l.x, B1l.y, B1l.z, B1l.w, B1h.x, B1h.y, B1h.z, B1h.w };
    float sbv[NF] = { SBl.x, SBl.y, SBl.z, SBl.w, SBh.x, SBh.y, SBh.z, SBh.w };
    float wn[NF] = { WNl.x, WNl.y, WNl.z, WNl.w, WNh.x, WNh.y, WNh.z, WNh.w };

    v8f out = {};   // weighted dissimilarity accumulator (16x16 C/D layout)

#pragma unroll
    for (int f = 0; f < NF; ++f) {
        v2f A, B;
        A.x = a0[f];
        A.y = a1[f] * pad;
        B.x = b0[f];
        B.y = b1[f] * pad;

        v8f acc = {};
        acc = __builtin_amdgcn_wmma_f32_16x16x4_f32(
            /*neg_a=*/false, A, /*neg_b=*/false, B,
            /*c_mod=*/(short)0, acc, /*reuse_a=*/false, /*reuse_b=*/false);

        const v4f* sap = (const v4f*)(sa + f * CPAD + cbase);
        v4f s0 = sap[0];
        v4f s1 = sap[1];
        float sav[8] = { s0.x, s0.y, s0.z, s0.w, s1.x, s1.y, s1.z, s1.w };

#pragma unroll
        for (int r = 0; r < 8; ++r) {
            float d2 = fmaf(-2.0f, acc[r], sav[r] + sbv[f]);
            d2 = fmaxf(d2, 0.0f);
            // raw v_sqrt_f32 (~1 ulp): plenty for top-k ordering, avoids the
            // IEEE-correct multi-instruction refinement sequence.
            out[r] = fmaf(wn[f], __builtin_amdgcn_sqrtf(d2), out[r]);
        }
    }

#pragma unroll
    for (int r = 0; r < 8; ++r) {
        int c = cbase + r;
        if (c < C_ROWS)                    // cols up to 9999 live inside stride
            D[c * HPAD + hn] = out[r];
    }
}

// ---------------------------------------------------------------------------
// Kernel 4: exact iterative top-50 (smallest) per row, ties -> lowest index.
// One block per row; D is L2-resident. out[c][k] = y_h[idx_k + 1].
// ---------------------------------------------------------------------------
__global__ void anen_topk(float* __restrict__ D,
                          const float* __restrict__ yh,
                          float* __restrict__ out) {
    __shared__ unsigned long long red[256];
    const int row = blockIdx.x;
    const int tid = threadIdx.x;
    float* Drow = D + (size_t)row * HPAD;

    for (int k = 0; k < KSEL; ++k) {
        unsigned long long best = ~0ULL;
        // vectorized scan: 4 consecutive floats per thread, 16B aligned;
        // cols 9998/9999 are in-bounds garbage -> masked out of keys.
        for (int j4 = tid * 4; j4 < H_ROWS; j4 += 1024) {
            v4f v = *(const v4f*)(Drow + j4);
            float vv[4] = { v.x, v.y, v.z, v.w };
#pragma unroll
            for (int jj = 0; jj < 4; ++jj) {
                int j = j4 + jj;
                if (j < H_ROWS) {
                    unsigned u = __float_as_uint(vv[jj]);
                    u = (u & 0x80000000u) ? ~u : (u | 0x80000000u);
                    unsigned long long key =
                        ((unsigned long long)u << 32) | (unsigned)j;
                    best = key < best ? key : best;
                }
            }
        }
        red[tid] = best;
        __syncthreads();
        for (int s = 128; s > 0; s >>= 1) {
            if (tid < s) {
                unsigned long long o = red[tid + s];
                if (o < red[tid]) red[tid] = o;
            }
            __syncthreads();
        }
        if (tid == 0) {
            unsigned idx = (unsigned)(red[0] & 0xffffffffu);
            out[row * KSEL + k] = yh[idx + 1];
            Drow[idx] = __builtin_inff();   // exclude from next iteration
            __threadfence();
        }
        __syncthreads();
    }
}

// ---------------------------------------------------------------------------
extern "C" void kernel_launch(void* const* d_in, const int* in_sizes, int n_in,
                              void* d_out, int out_size, void* d_ws, size_t ws_size,
                              hipStream_t stream) {
    const float* xt = (const float*)d_in[0];   // (512, 8)
    const float* xh = (const float*)d_in[1];   // (10000, 8)
    const float* yh = (const float*)d_in[2];   // (10000, 1)
    const float* fw = (const float*)d_in[3];   // (8,)
    float* ws   = (float*)d_ws;
    float* out  = (float*)d_out;               // (510, 50)

    float* wnorm = ws + WS_WNORM;
    float* sa    = ws + WS_SA;
    float* sbr   = ws + WS_SB;
    float* D     = ws + WS_D;

    anen_stats<<<1, 256, 0, stream>>>(xh, fw, wnorm);
    anen_wsq<<<(NF * CPAD + 255) / 256, 256, 0, stream>>>(xt, sa, 512, CPAD, 1);
    anen_wsq<<<(NF * HPAD + 255) / 256, 256, 0, stream>>>(xh, sbr, 10000, HPAD, 0);
    anen_dist_wmma<<<N_TILES / 4, 128, 0, stream>>>(xt, xh, wnorm, sa, sbr, D);
    anen_topk<<<C_ROWS, 256, 0, stream>>>(D, yh, out);
}